// MolecularGCNEncoder_78503412236483
// MI455X (gfx1250) — compile-verified
//
#include <hip/hip_runtime.h>

// ---------------------------------------------------------------------------
// MolecularGCNEncoder on gfx1250 (MI455X, wave32, WMMA)
//   xw = x @ W1                  (K=128)   -> stored pair-PACKED for WMMA-B
//   h  = adj @ xw + b1           (K=8192, streams 256MB adj, NT loads)
//   g  = h @ W2                  (K=32)    -> stored pair-PACKED
//   h2 = adj @ g  + b2           (K=8192, streams 256MB adj, NT loads)
//   out[32] = max over rows of h2
// fp32 end-to-end via V_WMMA_F32_16X16X4_F32 (matches fp32 reference).
// Memory floor: 2 x 256MB @ 23.3TB/s ~= 22us; kernel is built to stay there.
// ---------------------------------------------------------------------------

typedef __attribute__((ext_vector_type(2))) float v2f;
typedef __attribute__((ext_vector_type(8))) float v8f;

#define GN     8192                 // rows of adj / x
#define GH     32                   // hidden width (both layers)
#define TILEE  ((size_t)GN * GH)    // elements of one [8192 x 32] buffer

// Pair-packed B layout: PB[(r>>1)*64 + c*2 + (r&1)] = B[r][c]
// => the WMMA-B per-lane pair (B[kk,c], B[kk+1,c]) (kk even) is one v2f load.

// ---------------------------------------------------------------------------
// Tall-skinny GEMM partial for one 16-row tile x 32 cols, K-chunk blockIdx.y:
//   P[cy][rows][0..31] (+)= A[rows, cy*kChunk .. ) @ B
// PACK_B:     read B in pair-packed layout (xw / g)
// PACK_STORE: write C in pair-packed layout (producing xw / g)
// A loads are non-temporal (streaming adj must not thrash L2).
// ---------------------------------------------------------------------------
template <bool PACK_B, bool PACK_STORE>
__global__ __launch_bounds__(256)
void gcn_gemm16(const float* __restrict__ A, int lda,
                const float* __restrict__ Bm,
                float* __restrict__ P, int kChunk) {
    const int wave   = (int)(blockIdx.x * (blockDim.x >> 5)) + (threadIdx.x >> 5);
    const int lane   = threadIdx.x & 31;
    const int half   = lane >> 4;      // 0: K=0,1 / rows 0..7 ; 1: K=2,3 / rows 8..15
    const int ln     = lane & 15;
    const int kBegin = blockIdx.y * kChunk;

    float* __restrict__ C = P + (size_t)blockIdx.y * TILEE;

    // A: lane (half,ln) supplies A[row = wave*16+ln, kBegin+k+2*half .. +1]
    const float* __restrict__ arow =
        A + (size_t)(wave * 16 + ln) * (size_t)lda + (size_t)kBegin + 2 * half;

    // B base pointers for both layouts
    const float* __restrict__ brow_u =
        Bm + (size_t)(kBegin + 2 * half) * GH + ln;                 // unpacked
    const float* __restrict__ brow_p =
        Bm + ((size_t)(kBegin >> 1) + half) * (2 * GH) + 2 * ln;    // packed

    v8f acc0{};   // columns 0..15
    v8f acc1{};   // columns 16..31

#pragma unroll 8
    for (int k = 0; k < kChunk; k += 4) {
        if ((k & 31) == 0)
            __builtin_prefetch(arow + k + 256, 0, 3);   // global_prefetch_b8

        // streaming operand: non-temporal (th:TH_LOAD_NT)
        v2f a = __builtin_nontemporal_load((const v2f*)(arow + k));

        v2f b0, b1;
        if (PACK_B) {
            const float* bp = brow_p + (size_t)(k >> 1) * (2 * GH);
            b0 = *(const v2f*)(bp);            // cols 0..15 pair
            b1 = *(const v2f*)(bp + 2 * GH / 2 * 0 + 32);  // cols 16..31 pair
        } else {
            const float* bp = brow_u + (size_t)k * GH;
            b0[0] = bp[0];  b0[1] = bp[GH];
            b1[0] = bp[16]; b1[1] = bp[GH + 16];
        }

        acc0 = __builtin_amdgcn_wmma_f32_16x16x4_f32(false, a, false, b0,
                                                     (short)0, acc0, false, false);
        acc1 = __builtin_amdgcn_wmma_f32_16x16x4_f32(false, a, false, b1,
                                                     (short)0, acc1, false, false);
    }

    // C/D layout: lane (half,ln) holds col ln / 16+ln; vgpr j = row j+8*half.
    const int rowBase = wave * 16 + 8 * half;
#pragma unroll
    for (int j = 0; j < 8; ++j) {
        const int r = rowBase + j;
        if (PACK_STORE) {
            float* cr = C + (size_t)(r >> 1) * (2 * GH) + 2 * ln + (r & 1);
            cr[0]      = acc0[j];
            cr[2 * 16] = acc1[j];
        } else {
            float* cr = C + (size_t)r * GH + ln;
            cr[0]  = acc0[j];
            cr[16] = acc1[j];
        }
    }
}

// ---------------------------------------------------------------------------
// C[i] = bias[i%32] + sum_{c<nchunks} P[c][i]  (fixed order => deterministic).
// Safe in-place when P == C and nchunks == 1. Partials are read once: NT.
// ---------------------------------------------------------------------------
__global__ __launch_bounds__(256)
void gcn_reduce_bias(const float* __restrict__ P, int nchunks,
                     const float* __restrict__ bias, float* __restrict__ C) {
    const size_t i = (size_t)blockIdx.x * blockDim.x + threadIdx.x;   // < 8192*32
    float s = bias[i & (GH - 1)];
    for (int c = 0; c < nchunks; ++c)
        s += __builtin_nontemporal_load(P + (size_t)c * TILEE + i);
    C[i] = s;
}

// ---------------------------------------------------------------------------
// Exact, order-independent float atomic max (sign-aware int/uint trick).
// ---------------------------------------------------------------------------
__device__ __forceinline__ void atomic_max_f32(float* addr, float val) {
    if (!(val < 0.0f))
        atomicMax((int*)addr, __float_as_int(val));
    else
        atomicMin((unsigned int*)addr, __float_as_uint(val));
}

__global__ void gcn_init_out(float* __restrict__ out) {
    out[threadIdx.x] = -__builtin_inff();
}

__global__ __launch_bounds__(256)
void gcn_colmax(const float* __restrict__ H, float* __restrict__ out) {
    const int col     = threadIdx.x & (GH - 1);
    const int rstart  = blockIdx.x * (blockDim.x >> 5) + (threadIdx.x >> 5);
    const int rstride = gridDim.x * (blockDim.x >> 5);
    float m = -__builtin_inff();
    for (int r = rstart; r < GN; r += rstride)
        m = fmaxf(m, H[(size_t)r * GH + col]);
    atomic_max_f32(&out[col], m);
}

// ---------------------------------------------------------------------------
extern "C" void kernel_launch(void* const* d_in, const int* in_sizes, int n_in,
                              void* d_out, int out_size, void* d_ws, size_t ws_size,
                              hipStream_t stream) {
    (void)in_sizes; (void)n_in; (void)out_size;
    const float* x   = (const float*)d_in[0];   // [8192,128]
    const float* adj = (const float*)d_in[1];   // [8192,8192]
    const float* W1  = (const float*)d_in[2];   // [128,32]
    const float* b1  = (const float*)d_in[3];   // [32]
    const float* W2  = (const float*)d_in[4];   // [32,32]
    const float* b2  = (const float*)d_in[5];   // [32]
    float* out = (float*)d_out;                 // [32]

    float* xw    = (float*)d_ws;            // 1 MB, pair-packed [8192,32]
    float* h     = xw + TILEE;              // 1 MB, row-major
    float* g     = h  + TILEE;              // 1 MB, pair-packed
    float* pbase = g  + TILEE;              // K-split partials (if room)
    float* h2    = xw;                      // xw dead after step 2; reuse

    // Deterministic K-split factor from available scratch.
    size_t spare = ws_size > 3 * TILEE * sizeof(float)
                       ? (ws_size - 3 * TILEE * sizeof(float)) / (TILEE * sizeof(float))
                       : 0;
    int ksplit = spare >= 8 ? 8 : spare >= 4 ? 4 : spare >= 2 ? 2 : 1;

    const dim3 blk(256);                    // 8 waves; 1 wave per 16-row tile
    const dim3 grd1(GN / 16 / 8, 1);        // 64 blocks cover 512 row tiles

    // 1) xw = x @ W1   (unpacked B = W1, packed store)
    gcn_gemm16<false, true><<<grd1, blk, 0, stream>>>(x, 128, W1, xw, 128);

    // 2) h = adj @ xw + b1   (packed B = xw, standard partial store)
    {
        float* part = (ksplit > 1) ? pbase : h;
        gcn_gemm16<true, false><<<dim3(GN / 16 / 8, ksplit), blk, 0, stream>>>(
            adj, GN, xw, part, GN / ksplit);
        gcn_reduce_bias<<<(unsigned)(TILEE / 256), blk, 0, stream>>>(part, ksplit, b1, h);
    }

    // 3) g = h @ W2   (A = h row-major, unpacked B = W2, packed store)
    gcn_gemm16<false, true><<<grd1, blk, 0, stream>>>(h, GH, W2, g, GH);

    // 4) h2 = adj @ g + b2   (packed B = g)
    {
        float* part = (ksplit > 1) ? pbase : h2;
        gcn_gemm16<true, false><<<dim3(GN / 16 / 8, ksplit), blk, 0, stream>>>(
            adj, GN, g, part, GN / ksplit);
        gcn_reduce_bias<<<(unsigned)(TILEE / 256), blk, 0, stream>>>(part, ksplit, b2, h2);
    }

    // 5) out = max over rows of h2
    gcn_init_out<<<1, GH, 0, stream>>>(out);
    gcn_colmax<<<64, blk, 0, stream>>>(h2, out);
}